// UpsamplingNearestSingle_22359599743098
// MI455X (gfx1250) — compile-verified
//
#include <hip/hip_runtime.h>
#include <hip/hip_bf16.h>
#include <stdint.h>

// ---------------------------------------------------------------------------
// Nearest-neighbor 2x voxel subdivision (SCALE=2 -> 8 children per voxel).
//   fine_data[n*8+s, :] = coarse_data[n, :]          (64 f32 = 256 B per row)
//   fine_ijk [n*8+s, c] = 2*coarse_ijk[n, c] + bit   (meshgrid 'ij' order)
// Pure data movement, ~603 MiB HBM traffic -> bandwidth bound (~25 us floor).
// Fast path uses the CDNA5 async LDS data mover: each coarse row is staged in
// LDS once via GLOBAL_LOAD_ASYNC_TO_LDS_B128, then broadcast 8x to memory via
// GLOBAL_STORE_ASYNC_FROM_LDS_B128 (ASYNCcnt-tracked, no VGPR round trip).
// ---------------------------------------------------------------------------

__device__ __forceinline__ void async_load_b128(unsigned lds_addr,
                                                unsigned long long gaddr) {
  asm volatile("global_load_async_to_lds_b128 %0, %1, off"
               :
               : "v"(lds_addr), "v"(gaddr)
               : "memory");
}

__device__ __forceinline__ void async_store_b128(unsigned long long gaddr,
                                                 unsigned lds_addr) {
  asm volatile("global_store_async_from_lds_b128 %0, %1, off"
               :
               : "v"(gaddr), "v"(lds_addr)
               : "memory");
}

__device__ __forceinline__ void wait_asynccnt0() {
  asm volatile("s_wait_asynccnt 0x0" ::: "memory");
}

// ---------------------------------------------------------------------------
// fine_data fast path: C == 64 (256 B rows), groups of 8 coarse rows per wave
// iteration. Per iteration: 4 async loads (2 KiB in), 1 wait, 32 async stores
// (16 KiB out). LDS double buffer (2 x 2 KiB per wave) lets next iteration's
// loads overlap the 32 outstanding stores.
// ---------------------------------------------------------------------------
__global__ void __launch_bounds__(256)
upsample_data_async_kernel(const float* __restrict__ src,
                           float* __restrict__ dst,
                           int n_groups) {
  __shared__ __align__(16) char smem[8 * 4096];  // 8 waves x (2 x 2 KiB)

  const int lane = threadIdx.x & 31;
  const int wave = threadIdx.x >> 5;
  const int wavesPerBlock = blockDim.x >> 5;
  const int gw = blockIdx.x * wavesPerBlock + wave;
  const int stride = gridDim.x * wavesPerBlock;

  // Low 32 bits of the generic (flat) address of LDS == wave-relative LDS
  // offset (flat->LDS mapping truncates to addr[31:0]).
  const unsigned ldsWaveBase =
      (unsigned)(unsigned long long)(uintptr_t)(&smem[0]) + (unsigned)(wave * 4096);

  const unsigned long long srcB = (unsigned long long)(uintptr_t)src;
  const unsigned long long dstB = (unsigned long long)(uintptr_t)dst;
  const unsigned laneOff16 = (unsigned)(lane * 16);
  const unsigned laneWrap = laneOff16 & 255u;  // half-wave wraps onto one row

  unsigned it = 0;
  for (int g = gw; g < n_groups; g += stride, ++it) {
    const unsigned bufOff = ldsWaveBase + ((it & 1u) * 2048u);
    const unsigned long long srcBase = srcB + (unsigned long long)g * 2048ull;
    const unsigned long long dstBase = dstB + (unsigned long long)g * 16384ull;

    // Stage 8 coarse rows (2 KiB) into this wave's LDS buffer.
#pragma unroll
    for (int k = 0; k < 4; ++k) {
      async_load_b128(bufOff + (unsigned)(k * 512) + laneOff16,
                      srcBase + (unsigned long long)(k * 512) + laneOff16);
    }

    // Drain our loads (also drains the previous iteration's stores, which
    // read the *other* buffer, so cross-iteration overlap is preserved).
    wait_asynccnt0();

    // Fan out: each coarse row -> 8 contiguous 256 B copies (2 KiB block).
    // One 512 B store covers two copies (half-wave LDS address wrap).
#pragma unroll
    for (int r = 0; r < 8; ++r) {
      const unsigned ldsRow = bufOff + (unsigned)(r * 256) + laneWrap;
      const unsigned long long memRow =
          dstBase + (unsigned long long)(r * 2048) + laneOff16;
#pragma unroll
      for (int t = 0; t < 4; ++t) {
        async_store_b128(memRow + (unsigned long long)(t * 512), ldsRow);
      }
    }
  }
  // S_ENDPGM performs an implicit wait-idle, draining outstanding stores.
}

// ---------------------------------------------------------------------------
// Generic fallback for unexpected shapes (C != 64 or N % 8 != 0).
// ---------------------------------------------------------------------------
__global__ void upsample_data_generic_kernel(const float* __restrict__ src,
                                             float* __restrict__ dst,
                                             long long n_vox, int C) {
  const long long total = n_vox * 8ll * (long long)C;
  const long long step = (long long)gridDim.x * blockDim.x;
  for (long long i = (long long)blockIdx.x * blockDim.x + threadIdx.x;
       i < total; i += step) {
    const long long f = i / C;         // fine row
    const int c = (int)(i - f * C);    // channel
    const long long n = f >> 3;        // coarse row
    dst[i] = src[n * C + c];
  }
}

// ---------------------------------------------------------------------------
// fine_ijk: [N*8, 3] int32.  offsets (meshgrid 'ij'): comp c bit = (s>>(2-c))&1
// ---------------------------------------------------------------------------
__global__ void upsample_ijk_kernel(const int* __restrict__ ijk,
                                    int* __restrict__ out, int n_fine) {
  const int f = blockIdx.x * blockDim.x + threadIdx.x;
  if (f >= n_fine) return;
  const int n = f >> 3;
  const int s = f & 7;
  const int b = n * 3;
  const int o = f * 3;
  out[o + 0] = (ijk[b + 0] << 1) + ((s >> 2) & 1);
  out[o + 1] = (ijk[b + 1] << 1) + ((s >> 1) & 1);
  out[o + 2] = (ijk[b + 2] << 1) + (s & 1);
}

extern "C" void kernel_launch(void* const* d_in, const int* in_sizes, int n_in,
                              void* d_out, int out_size, void* d_ws,
                              size_t ws_size, hipStream_t stream) {
  const float* coarse_data = (const float*)d_in[0];
  const int* coarse_ijk = (const int*)d_in[1];

  const long long n_vox = (long long)in_sizes[1] / 3;        // 262144
  const int C = (int)((long long)in_sizes[0] / n_vox);       // 64
  const long long n_fine = n_vox * 8ll;                      // 2097152

  float* fine_data = (float*)d_out;
  int* fine_ijk = (int*)(fine_data + n_fine * (long long)C);

  if (C == 64 && (n_vox % 8) == 0) {
    const int n_groups = (int)(n_vox / 8);                   // 32768
    int blocks = 1024;                                       // 8192 waves
    const int waves = blocks * 8;
    if (waves > n_groups) blocks = (n_groups + 7) / 8;
    if (blocks < 1) blocks = 1;
    upsample_data_async_kernel<<<blocks, 256, 0, stream>>>(coarse_data,
                                                           fine_data, n_groups);
  } else {
    upsample_data_generic_kernel<<<4096, 256, 0, stream>>>(coarse_data,
                                                           fine_data, n_vox, C);
  }

  const int ijk_blocks = (int)((n_fine + 255) / 256);
  upsample_ijk_kernel<<<ijk_blocks, 256, 0, stream>>>(coarse_ijk, fine_ijk,
                                                      (int)n_fine);
}